// DeformConv_50912542326879
// MI455X (gfx1250) — compile-verified
//
#include <hip/hip_runtime.h>
#include <math.h>

// ---------------------------------------------------------------------------
// DCNv2 block: offset/mask conv -> modulated deformable conv (WMMA f32 GEMM)
//              -> concat 1x1 conv + BN + SiLU
// Shapes: B=8, C1=C2=128, H=W=64, K=3 (KK=9), stride=1, pad=1, dil=1
// ---------------------------------------------------------------------------

#define BB 8
#define C1 128
#define C2 128
#define HH 64
#define WW 64
#define HW 4096
#define KKT 9
#define OMCH 27          // 18 offset + 9 mask channels
#define SSTR 132         // padded LDS row stride (dwords): conflict-free b64

typedef float v2f __attribute__((ext_vector_type(2)));
typedef float v8f __attribute__((ext_vector_type(8)));

// ---------------------------------------------------------------------------
// Kernel 0: transpose deform weights w_def[o][ci][kk] -> wT[kk][o][ci]
// so WMMA A-fragments (two consecutive ci for one o) load as one b64.
// ---------------------------------------------------------------------------
__global__ __launch_bounds__(256) void k_wt(const float* __restrict__ w_def,
                                            float* __restrict__ wT)
{
  int e = blockIdx.x * 256 + threadIdx.x;   // 0 .. 9*128*128-1
  int ci = e & 127;
  int o  = (e >> 7) & 127;
  int kk = e >> 14;
  wT[e] = w_def[((o << 7) + ci) * 9 + kk];
}

// ---------------------------------------------------------------------------
// Kernel 1: 3x3 conv producing offset[b,18,h,w] and sigmoid(mask)[b,9,h,w]
// One block per (b, row). 256 threads: pixel = t&63, channel-group = t>>6.
// ---------------------------------------------------------------------------
__global__ __launch_bounds__(256) void k_offmask(
    const float* __restrict__ x,
    const float* __restrict__ w_off, const float* __restrict__ b_off,
    const float* __restrict__ w_msk, const float* __restrict__ b_msk,
    float* __restrict__ offb, float* __restrict__ maskb)
{
  __shared__ float srow[3][66];        // 3 input rows, cols -1..64 (zero-pad)
  __shared__ float swci[OMCH * 9];     // 27 channels x 9 taps for current ci

  const int blk = blockIdx.x;          // 0..511
  const int b = blk >> 6;
  const int y = blk & 63;
  const int t = threadIdx.x;
  const int px = t & 63;
  const int cg = t >> 6;               // 0..3

  float acc[7];
#pragma unroll
  for (int j = 0; j < 7; ++j) acc[j] = 0.f;

  for (int ci = 0; ci < C1; ++ci) {
    __syncthreads();                   // protect LDS from previous iteration
    for (int e = t; e < 198; e += 256) {
      int r = e / 66, cc = e - r * 66;
      int yy = y + r - 1, xx = cc - 1;
      float v = 0.f;
      if (yy >= 0 && yy < HH && xx >= 0 && xx < WW)
        v = x[(((b * C1) + ci) << 12) + (yy << 6) + xx];
      srow[r][cc] = v;
    }
    for (int e = t; e < OMCH * 9; e += 256) {
      int c = e / 9, tap = e - c * 9;
      swci[e] = (c < 18) ? w_off[(c * C1 + ci) * 9 + tap]
                         : w_msk[((c - 18) * C1 + ci) * 9 + tap];
    }
    __syncthreads();

    float xv[9];
#pragma unroll
    for (int tap = 0; tap < 9; ++tap) {
      int r = tap / 3, cc = tap - r * 3;
      xv[tap] = srow[r][px + cc];
    }
#pragma unroll
    for (int j = 0; j < 7; ++j) {
      int c = cg + 4 * j;
      if (c < OMCH) {
        float a = acc[j];
#pragma unroll
        for (int tap = 0; tap < 9; ++tap)
          a = fmaf(swci[c * 9 + tap], xv[tap], a);
        acc[j] = a;
      }
    }
  }

#pragma unroll
  for (int j = 0; j < 7; ++j) {
    int c = cg + 4 * j;
    if (c < OMCH) {
      if (c < 18) {
        offb[((b * 18 + c) << 12) + (y << 6) + px] = acc[j] + b_off[c];
      } else {
        float z = acc[j] + b_msk[c - 18];
        maskb[((b * 9 + (c - 18)) << 12) + (y << 6) + px] =
            1.f / (1.f + expf(-z));
      }
    }
  }
}

// ---------------------------------------------------------------------------
// Kernel 2: modulated deformable conv as WMMA f32 GEMM.
//   D[C2=128, pix] = Wdef[128, K=1152] * Val[1152, pix]
// K ordering: k = kk*128 + ci (tap-major). One block = one full 64-pixel row,
// 8 waves; wave w owns output rows 16w..16w+15 and four N-tiles of 16.
// Per tap: gather/modulate transposed slab sT[n][ci] into LDS, then 32 K=4
// steps: one b64 A load + four b64 LDS B loads feeding 4 WMMAs.
// ---------------------------------------------------------------------------
__global__ __launch_bounds__(256) void k_deform(
    const float* __restrict__ x,
    const float* __restrict__ offb, const float* __restrict__ maskb,
    const float* __restrict__ wT, float* __restrict__ xd)
{
  __shared__ int   sIdx[KKT][64][4];   // clamped corner plane-indices
  __shared__ float sWgt[KKT][64][4];   // bilinear weight * mask (0 if OOB)
  __shared__ float sT[64][SSTR];       // sampled slab, pixel-major (+pad)

  const int blk = blockIdx.x;          // 0..511
  const int b   = blk >> 6;
  const int y   = blk & 63;
  const int t   = threadIdx.x;

  // ---- phase 1: sampling metadata for 9 taps x 64 pixels ----
  for (int e = t; e < KKT * 64; e += 256) {
    int kk = e >> 6;
    int n  = e & 63;
    int obase = ((b * 18 + 2 * kk) << 12) + (y << 6) + n;
    float dy = offb[obase];
    float dx = offb[obase + HW];
    float m  = maskb[((b * 9 + kk) << 12) + (y << 6) + n];
    int ti = kk / 3, tj = kk - ti * 3;
    float py  = dy + (float)(y - 1 + ti);
    float pxs = dx + (float)(n - 1 + tj);
    float y0f = floorf(py), x0f = floorf(pxs);
    float wy1 = py - y0f, wx1 = pxs - x0f;
    float wy[2] = {1.f - wy1, wy1};
    float wx[2] = {1.f - wx1, wx1};
    int iy0 = (int)y0f, ix0 = (int)x0f;
#pragma unroll
    for (int c = 0; c < 4; ++c) {
      int ry = iy0 + (c >> 1);
      int rx = ix0 + (c & 1);
      bool ok = (ry >= 0) && (ry < HH) && (rx >= 0) && (rx < WW);
      sIdx[kk][n][c] = ok ? ((ry << 6) + rx) : 0;
      sWgt[kk][n][c] = ok ? (wy[c >> 1] * wx[c & 1] * m) : 0.f;
    }
  }
  __syncthreads();

  const int wid  = t >> 5;             // wave id: M-tile (rows 16*wid..)
  const int lane = t & 31;
  const int lm   = lane & 15;          // A: M row / B: N col within tile
  const int lh   = lane >> 4;          // selects K pair {2*lh, 2*lh+1}
  const int xplane = (b * C1) << 12;

  v8f acc0 = {}, acc1 = {}, acc2 = {}, acc3 = {};

  for (int kk = 0; kk < KKT; ++kk) {
    // ---- gather + modulate slab sT[n][ci] (lanes share ci: good locality) --
    for (int e = t; e < C1 * 64; e += 256) {
      int ci = e >> 6, n = e & 63;
      const int*   id = sIdx[kk][n];
      const float* wt = sWgt[kk][n];
      const float* xp = x + xplane + (ci << 12);
      sT[n][ci] = wt[0] * xp[id[0]] + wt[1] * xp[id[1]] +
                  wt[2] * xp[id[2]] + wt[3] * xp[id[3]];
    }
    __syncthreads();

    // ---- K=128 sub-GEMM via V_WMMA_F32_16X16X4_F32 ----
    const int o = (wid << 4) + lm;
    const float* wk = wT + ((kk * C2 + o) << 7);   // 128 contiguous ci
    for (int ci0 = 0; ci0 < C1; ci0 += 4) {
      int kA = ci0 + (lh << 1);
      v2f af = *(const v2f*)(wk + kA);             // one global_load_b64
      v2f b0 = *(const v2f*)(&sT[lm][kA]);         // one ds_load_b64 each
      v2f b1 = *(const v2f*)(&sT[16 + lm][kA]);
      v2f b2 = *(const v2f*)(&sT[32 + lm][kA]);
      v2f b3 = *(const v2f*)(&sT[48 + lm][kA]);
      acc0 = __builtin_amdgcn_wmma_f32_16x16x4_f32(
          false, af, false, b0, (short)0, acc0, false, false);
      acc1 = __builtin_amdgcn_wmma_f32_16x16x4_f32(
          false, af, false, b1, (short)0, acc1, false, false);
      acc2 = __builtin_amdgcn_wmma_f32_16x16x4_f32(
          false, af, false, b2, (short)0, acc2, false, false);
      acc3 = __builtin_amdgcn_wmma_f32_16x16x4_f32(
          false, af, false, b3, (short)0, acc3, false, false);
    }
    __syncthreads();
  }

  // ---- store D tiles: VGPR v -> row M = v + 8*lh, col N = lm ----
#pragma unroll
  for (int v = 0; v < 8; ++v) {
    int oo = (wid << 4) + v + (lh << 3);
    int outbase = ((b * C2 + oo) << 12) + (y << 6);
    xd[outbase + lm]      = acc0[v];
    xd[outbase + 16 + lm] = acc1[v];
    xd[outbase + 32 + lm] = acc2[v];
    xd[outbase + 48 + lm] = acc3[v];
  }
}

// ---------------------------------------------------------------------------
// Kernel 3: 1x1 conv over concat[xd(128), offset(18), mask(9)] + BN + SiLU.
// One block per (b,row); cat row staged in LDS; weight loads wave-uniform.
// ---------------------------------------------------------------------------
__global__ __launch_bounds__(256) void k_post(
    const float* __restrict__ xd,
    const float* __restrict__ offb, const float* __restrict__ maskb,
    const float* __restrict__ w1, const float* __restrict__ b1,
    const float* __restrict__ gamma, const float* __restrict__ beta,
    const float* __restrict__ mean, const float* __restrict__ var,
    float* __restrict__ out)
{
  __shared__ float scat[155 * 64];     // 38.75 KB

  const int blk = blockIdx.x;          // 0..511
  const int b = blk >> 6;
  const int y = blk & 63;
  const int t = threadIdx.x;

  for (int e = t; e < 155 * 64; e += 256) {
    int c = e >> 6, px = e & 63;
    float v;
    if (c < 128)
      v = xd[((b * C2 + c) << 12) + (y << 6) + px];
    else if (c < 146)
      v = offb[((b * 18 + (c - 128)) << 12) + (y << 6) + px];
    else
      v = maskb[((b * 9 + (c - 146)) << 12) + (y << 6) + px];
    scat[(c << 6) + px] = v;
  }
  __syncthreads();

  const int px = t & 63;
  const int og = t >> 6;               // 0..3

  for (int jb = 0; jb < 4; ++jb) {
    float acc[8];
#pragma unroll
    for (int jj = 0; jj < 8; ++jj) acc[jj] = 0.f;
    for (int i = 0; i < 155; ++i) {
      float v = scat[(i << 6) + px];
#pragma unroll
      for (int jj = 0; jj < 8; ++jj) {
        int o = og + 4 * (jb * 8 + jj);
        acc[jj] = fmaf(w1[o * 155 + i], v, acc[jj]);
      }
    }
#pragma unroll
    for (int jj = 0; jj < 8; ++jj) {
      int o = og + 4 * (jb * 8 + jj);
      float inv = gamma[o] / sqrtf(var[o] + 1e-5f);
      float z = acc[jj] + b1[o];
      z = z * inv + (beta[o] - mean[o] * inv);
      out[((b * C2 + o) << 12) + (y << 6) + px] = z / (1.f + expf(-z));
    }
  }
}

// ---------------------------------------------------------------------------
extern "C" void kernel_launch(void* const* d_in, const int* in_sizes, int n_in,
                              void* d_out, int out_size, void* d_ws,
                              size_t ws_size, hipStream_t stream) {
  const float* x      = (const float*)d_in[0];
  const float* w_off  = (const float*)d_in[1];
  const float* b_off  = (const float*)d_in[2];
  const float* w_msk  = (const float*)d_in[3];
  const float* b_msk  = (const float*)d_in[4];
  const float* w_def  = (const float*)d_in[5];
  const float* w_1x1  = (const float*)d_in[6];
  const float* b_1x1  = (const float*)d_in[7];
  const float* gamma  = (const float*)d_in[8];
  const float* beta   = (const float*)d_in[9];
  const float* mean   = (const float*)d_in[10];
  const float* var    = (const float*)d_in[11];

  float* ws    = (float*)d_ws;
  float* offb  = ws;                              // 8*18*4096  floats
  float* maskb = offb + BB * 18 * HW;             // 8*9*4096   floats
  float* xdbuf = maskb + BB * 9 * HW;             // 8*128*4096 floats
  float* wTbuf = xdbuf + BB * C2 * HW;            // 9*128*128  floats

  k_wt<<<(KKT * C2 * C1) / 256, 256, 0, stream>>>(w_def, wTbuf);
  k_offmask<<<BB * HH, 256, 0, stream>>>(x, w_off, b_off, w_msk, b_msk,
                                         offb, maskb);
  k_deform<<<BB * HH, 256, 0, stream>>>(x, offb, maskb, wTbuf, xdbuf);
  k_post<<<BB * HH, 256, 0, stream>>>(xdbuf, offb, maskb, w_1x1, b_1x1,
                                      gamma, beta, mean, var, (float*)d_out);
}